// CSCDVTV_37692632990255
// MI455X (gfx1250) — compile-verified
//
#include <hip/hip_runtime.h>
#include <cmath>

#define MBATCH 4
#define CCH    3
#define MM     192
#define NNW    192
#define PP     32
#define KITER  8
#define KSZ    7
#define WCHW   0.8f
#define EPSI   1e-8f

typedef float v2f __attribute__((ext_vector_type(2)));
typedef float v8f __attribute__((ext_vector_type(8)));

// CMAT (opponent color transform)
__device__ __constant__ float d_CM[3][3] = {
    { 0.57735027f,  0.57735027f,  0.57735027f },
    { 0.70710678f, -0.70710678f,  0.0f        },
    { 0.40824829f,  0.40824829f, -0.81649658f }
};

// Compile-time LDS offset for tap k in the r halo tile (14 x 22):
// element for tap k at output pixel (wave,pl) lives at
//   (wave + 6 - k/7)*22 + (pl + 6 - k%7)  =  wave*22 + pl + OFF(k)
// Taps 49..51 are zero-weight padding; clamp their index in-tile.
__host__ __device__ constexpr int OFFT(int k) {
    int kk = (k < 49) ? k : 48;
    return (6 - kk / 7) * 22 + (6 - kk % 7);
}

// ---------------------------------------------------------------------------
// Kernel 1: r = x_bef - conv_csc(a, B)
// Degenerate GEMM (1 output channel/group) -> LDS-tiled VALU FMA.
// grid (12,12,12): (n-tile, m-tile, b*C). 256 threads = 16x16 pixel tile.
// ---------------------------------------------------------------------------
__global__ __launch_bounds__(256) void k_conv_csc(
    const float* __restrict__ a, const float* __restrict__ Bw,
    const float* __restrict__ xb, float* __restrict__ r)
{
    const int tid = threadIdx.x;
    const int b = blockIdx.z / CCH, c = blockIdx.z % CCH;
    const int m0 = blockIdx.y * 16, n0 = blockIdx.x * 16;
    const int tx = tid & 15, ty = tid >> 4;

    __shared__ float sw[PP * 49];
    __shared__ float sa[22 * 22];

    for (int i = tid; i < PP * 49; i += 256) {
        int p = i / 49, t = i % 49;
        sw[i] = Bw[(p * CCH + c) * 49 + t];
    }
    __syncthreads();

    float acc = 0.0f;
    for (int p = 0; p < PP; ++p) {
        const float* ap = a + (((size_t)(b * PP + p) * CCH + c) * MM) * NNW;
        for (int i = tid; i < 22 * 22; i += 256) {
            int ii = i / 22, jj = i % 22;
            int mg = m0 - 3 + ii, ng = n0 - 3 + jj;
            bool inb = (mg >= 0) && (mg < MM) && (ng >= 0) && (ng < NNW);
            sa[i] = inb ? ap[mg * NNW + ng] : 0.0f;
        }
        if (p + 1 < PP) {
            const float* apn = a + (((size_t)(b * PP + p + 1) * CCH + c) * MM) * NNW;
            __builtin_prefetch(apn + m0 * NNW + n0, 0, 0);   // global_prefetch_b8
        }
        __syncthreads();
        const float* swp = sw + p * 49;
        #pragma unroll
        for (int dy = 0; dy < 7; ++dy) {
            #pragma unroll
            for (int dx = 0; dx < 7; ++dx) {
                acc = fmaf(sa[(ty + dy) * 22 + (tx + dx)], swp[dy * 7 + dx], acc);
            }
        }
        __syncthreads();
    }
    int idx = ((b * CCH + c) * MM + (m0 + ty)) * NNW + (n0 + tx);
    r[idx] = xb[idx] - acc;
}

// ---------------------------------------------------------------------------
// Kernel 2: x_aft = clip(x_bef - g1*(r + Ctop(Dtop(y1)) + mask*y2)),
//           u = Cop(2*x_aft - x_bef).   One thread per (b,m,n), 3 channels.
// ---------------------------------------------------------------------------
__global__ __launch_bounds__(256) void k_xupd(
    const float* __restrict__ xb, const float* __restrict__ r,
    const float* __restrict__ y1, const float* __restrict__ y2,
    const float* __restrict__ mask, const float* __restrict__ gam1, int k,
    float* __restrict__ xa, float* __restrict__ u)
{
    int gid = blockIdx.x * 256 + threadIdx.x;
    int b = gid / (MM * NNW);
    int rem = gid % (MM * NNW);
    int m = rem / NNW, n = rem % NNW;
    float g1 = gam1[k];

    float w[3], vv[3];
    #pragma unroll
    for (int c = 0; c < 3; ++c) {
        int i0 = (((b * 2 + 0) * CCH + c) * MM + m) * NNW + n;
        int i1 = (((b * 2 + 1) * CCH + c) * MM + m) * NNW + n;
        float yv  = (m == MM - 1) ? 0.0f : y1[i0];
        float yvm = (m >= 1)      ? y1[i0 - NNW] : 0.0f;
        float yh  = (n == NNW - 1) ? 0.0f : y1[i1];
        float yhm = (n >= 1)       ? y1[i1 - 1] : 0.0f;
        w[c] = (yvm - yv) + (yhm - yh);
    }
    #pragma unroll
    for (int c = 0; c < 3; ++c) {
        // Ctop: t[c] = sum_cc CM[cc][c] * w[cc]
        float t = d_CM[0][c] * w[0] + d_CM[1][c] * w[1] + d_CM[2][c] * w[2];
        int idx = ((b * CCH + c) * MM + m) * NNW + n;
        float x = xb[idx] - g1 * (r[idx] + t + mask[idx] * y2[idx]);
        float xc = fminf(fmaxf(x, 0.0f), 1.0f);
        xa[idx] = xc;
        vv[c] = 2.0f * xc - xb[idx];
    }
    #pragma unroll
    for (int d = 0; d < 3; ++d) {
        float ud = d_CM[d][0] * vv[0] + d_CM[d][1] * vv[1] + d_CM[d][2] * vv[2];
        u[((b * CCH + d) * MM + m) * NNW + n] = ud;
    }
}

// ---------------------------------------------------------------------------
// Kernel 3: a = soft(a + g2*convT_csc(r,B), g2*lam1), in place, via WMMA f32.
// Per (b,c): out[32 p][pix] = W[32x49] x patches[49 x pix].
// grid (12,24,12): (n-tile/16, m-tile/8, b*C). 8 waves, one row each; each wave
// does two 16x16 f32 WMMA tiles (p 0-15, 16-31) over 13 K=4 steps (49 taps
// padded to 52 with zero weights). All LDS tap offsets fold to immediates.
// ---------------------------------------------------------------------------
__global__ __launch_bounds__(256) void k_convT_wmma(
    const float* __restrict__ r, const float* __restrict__ Bw,
    float* __restrict__ a,
    const float* __restrict__ gam2, const float* __restrict__ lam1, int k)
{
    const int tid = threadIdx.x;
    const int b = blockIdx.z / CCH, c = blockIdx.z % CCH;
    const int m0 = blockIdx.y * 8, n0 = blockIdx.x * 16;

    __shared__ float sw[PP * 52];   // weights, taps padded 49->52 with zeros
    __shared__ float sr[14 * 22];   // r halo tile: rows m0-3..m0+10, cols n0-3..n0+18

    for (int i = tid; i < PP * 52; i += 256) {
        int p = i / 52, t = i % 52;
        sw[i] = (t < 49) ? Bw[(p * CCH + c) * 49 + t] : 0.0f;
    }
    const float* rp = r + ((size_t)(b * CCH + c) * MM) * NNW;
    for (int i = tid; i < 14 * 22; i += 256) {
        int ii = i / 22, jj = i % 22;
        int mg = m0 - 3 + ii, ng = n0 - 3 + jj;
        bool inb = (mg >= 0) && (mg < MM) && (ng >= 0) && (ng < NNW);
        sr[i] = inb ? rp[mg * NNW + ng] : 0.0f;
    }
    __syncthreads();

    const int wave = tid >> 5, lane = tid & 31;
    const int pl = lane & 15, kh = lane >> 4;
    const int m = m0 + wave;
    const int base = wave * 22 + pl;          // runtime part of B-frag address
    const int awlo = pl * 52 + 2 * kh;        // A-frag row base (p 0..15)
    const int awhi = (pl + 16) * 52 + 2 * kh; // A-frag row base (p 16..31)

    v8f acc0 = {0.f, 0.f, 0.f, 0.f, 0.f, 0.f, 0.f, 0.f};
    v8f acc1 = {0.f, 0.f, 0.f, 0.f, 0.f, 0.f, 0.f, 0.f};

    #pragma unroll
    for (int t4 = 0; t4 < 13; ++t4) {
        // A fragment (16 p x 4 taps): lane pl holds taps {4t4+2kh, +1} of row p
        v2f alo, ahi, bf;
        alo.x = sw[awlo + 4 * t4];
        alo.y = sw[awlo + 4 * t4 + 1];
        ahi.x = sw[awhi + 4 * t4];
        ahi.y = sw[awhi + 4 * t4 + 1];
        // B fragment (4 taps x 16 pixels): offsets are compile-time immediates
        // selected by kh (lanes 0-15 -> taps {4t4,4t4+1}; 16-31 -> {4t4+2,+3}).
        const int o0 = kh ? OFFT(4 * t4 + 2) : OFFT(4 * t4);
        const int o1 = kh ? OFFT(4 * t4 + 3) : OFFT(4 * t4 + 1);
        bf.x = sr[base + o0];
        bf.y = sr[base + o1];

        acc0 = __builtin_amdgcn_wmma_f32_16x16x4_f32(false, alo, false, bf,
                                                     (short)0, acc0, false, false);
        acc1 = __builtin_amdgcn_wmma_f32_16x16x4_f32(false, ahi, false, bf,
                                                     (short)0, acc1, false, false);
    }

    // Epilogue: fused a update + soft threshold, in place.
    float g2 = gam2[k];
    float thr = g2 * lam1[k];
    int n = n0 + pl;
    #pragma unroll
    for (int i = 0; i < 8; ++i) {
        int p0 = i + 8 * kh;        // acc0 -> p in [0,16)
        int p1 = 16 + i + 8 * kh;   // acc1 -> p in [16,32)
        size_t idx0 = (((size_t)(b * PP + p0) * CCH + c) * MM + m) * NNW + n;
        size_t idx1 = (((size_t)(b * PP + p1) * CCH + c) * MM + m) * NNW + n;
        float an0 = fmaf(g2, acc0[i], a[idx0]);
        float an1 = fmaf(g2, acc1[i], a[idx1]);
        float s0 = fabsf(an0) - thr;
        float s1 = fabsf(an1) - thr;
        a[idx0] = (s0 > 0.0f) ? copysignf(s0, an0) : 0.0f;
        a[idx1] = (s1 > 0.0f) ? copysignf(s1, an1) : 0.0f;
    }
}

// ---------------------------------------------------------------------------
// Kernel 4: y1 = y1 + g3*Dop(u); y1 -= g3*prox_dvtv(y1/(g3+eps),...). In place.
// One thread per (b,m,n): 2 dirs x 3 channels.
// ---------------------------------------------------------------------------
__global__ __launch_bounds__(256) void k_y1(
    float* __restrict__ y1, const float* __restrict__ u,
    const float* __restrict__ gam3, const float* __restrict__ lam2, int k)
{
    int gid = blockIdx.x * 256 + threadIdx.x;
    int b = gid / (MM * NNW);
    int rem = gid % (MM * NNW);
    int m = rem / NNW, n = rem % NNW;
    float g3 = gam3[k];
    float ge = g3 + EPSI;
    float tt = lam2[k] / ge;

    float yt[2][3], s[2][3];
    #pragma unroll
    for (int c = 0; c < 3; ++c) {
        int ub = ((b * CCH + c) * MM + m) * NNW + n;
        float uc = u[ub];
        float dv = (m < MM - 1)  ? (u[ub + NNW] - uc) : 0.0f;
        float dh = (n < NNW - 1) ? (u[ub + 1]   - uc) : 0.0f;
        int i0 = (((b * 2 + 0) * CCH + c) * MM + m) * NNW + n;
        int i1 = (((b * 2 + 1) * CCH + c) * MM + m) * NNW + n;
        yt[0][c] = fmaf(g3, dv, y1[i0]);
        yt[1][c] = fmaf(g3, dh, y1[i1]);
        s[0][c] = yt[0][c] / ge;
        s[1][c] = yt[1][c] / ge;
    }
    float nl = sqrtf(s[0][0] * s[0][0] + s[1][0] * s[1][0]);
    float nc = sqrtf(s[0][1] * s[0][1] + s[0][2] * s[0][2] +
                     s[1][1] * s[1][1] + s[1][2] * s[1][2]);
    float sl = fmaxf(0.0f, 1.0f - tt / fmaxf(nl, 1e-12f));
    float sc = fmaxf(0.0f, 1.0f - tt * WCHW / fmaxf(nc, 1e-12f));
    #pragma unroll
    for (int d = 0; d < 2; ++d) {
        #pragma unroll
        for (int c = 0; c < 3; ++c) {
            float sca = (c == 0) ? sl : sc;
            int i = (((b * 2 + d) * CCH + c) * MM + m) * NNW + n;
            y1[i] = yt[d][c] - g3 * (s[d][c] * sca);
        }
    }
}

// ---------------------------------------------------------------------------
// Kernel 5: y2 += g3*mask*(2*xa - xb) (in place); partial per-block sums of
// (y2/(g3+eps) - z)^2. Each 256-thread block maps entirely inside one batch b.
// ---------------------------------------------------------------------------
__global__ __launch_bounds__(256) void k_y2acc(
    float* __restrict__ y2, const float* __restrict__ mask,
    const float* __restrict__ xa, const float* __restrict__ xb,
    const float* __restrict__ z, const float* __restrict__ gam3, int k,
    float* __restrict__ part)
{
    int tid = threadIdx.x;
    int gid = blockIdx.x * 256 + tid;
    float g3 = gam3[k];
    float ge = g3 + EPSI;
    float v = 2.0f * xa[gid] - xb[gid];
    float yn = fmaf(g3 * mask[gid], v, y2[gid]);
    y2[gid] = yn;
    float d = yn / ge - z[gid];
    __shared__ float sd[256];
    sd[tid] = d * d;
    __syncthreads();
    #pragma unroll
    for (int sft = 128; sft > 0; sft >>= 1) {
        if (tid < sft) sd[tid] += sd[tid + sft];
        __syncthreads();
    }
    if (tid == 0) part[blockIdx.x] = sd[0];
}

// Kernel 6: reduce 432 partials per batch -> projection scale factor per b.
__global__ __launch_bounds__(256) void k_y2red(
    const float* __restrict__ part, float* __restrict__ fac, float radius)
{
    int b = blockIdx.x, tid = threadIdx.x;
    const float* pb = part + b * 432;
    float s = pb[tid] + ((tid < 176) ? pb[256 + tid] : 0.0f);
    __shared__ float sd[256];
    sd[tid] = s;
    __syncthreads();
    #pragma unroll
    for (int sft = 128; sft > 0; sft >>= 1) {
        if (tid < sft) sd[tid] += sd[tid + sft];
        __syncthreads();
    }
    if (tid == 0) {
        float nb = sqrtf(sd[0]);
        fac[b] = fminf(1.0f, radius / fmaxf(nb, 1e-12f));
    }
}

// Kernel 7: y2 -= g3 * (z + (y2/(g3+eps) - z) * fac[b]). In place.
__global__ __launch_bounds__(256) void k_y2app(
    float* __restrict__ y2, const float* __restrict__ z,
    const float* __restrict__ fac, const float* __restrict__ gam3, int k)
{
    int gid = blockIdx.x * 256 + threadIdx.x;
    int b = gid / (CCH * MM * NNW);
    float g3 = gam3[k];
    float ge = g3 + EPSI;
    float d = y2[gid] / ge - z[gid];
    float pr = fmaf(d, fac[b], z[gid]);
    y2[gid] = y2[gid] - g3 * pr;
}

// ---------------------------------------------------------------------------
extern "C" void kernel_launch(void* const* d_in, const int* in_sizes, int n_in,
                              void* d_out, int out_size, void* d_ws, size_t ws_size,
                              hipStream_t stream)
{
    const float* z    = (const float*)d_in[0];
    const float* mask = (const float*)d_in[1];
    const float* a0   = (const float*)d_in[2];
    const float* Bw   = (const float*)d_in[3];
    const float* lam1 = (const float*)d_in[4];
    const float* lam2 = (const float*)d_in[5];
    const float* gam1 = (const float*)d_in[6];
    const float* gam2 = (const float*)d_in[7];
    const float* gam3 = (const float*)d_in[8];

    const size_t NIMG = (size_t)MBATCH * CCH * MM * NNW;          // 442368
    const size_t NA   = (size_t)MBATCH * PP * CCH * MM * NNW;     // 14155776
    const size_t NY1  = 2 * NIMG;

    float* W = (float*)d_ws;
    size_t o = 0;
    float* x0 = W + o;  o += NIMG;
    float* x1 = W + o;  o += NIMG;
    float* a  = W + o;  o += NA;
    float* y1 = W + o;  o += NY1;
    float* y2 = W + o;  o += NIMG;
    float* r  = W + o;  o += NIMG;
    float* u  = W + o;  o += NIMG;
    float* part = W + o; o += 1728;
    float* fac  = W + o; o += 4;

    const float radius = 0.02f * sqrtf((float)(CCH * MM * NNW));

    hipMemcpyAsync(x0, z, NIMG * sizeof(float), hipMemcpyDeviceToDevice, stream);
    hipMemcpyAsync(a, a0, NA * sizeof(float), hipMemcpyDeviceToDevice, stream);
    hipMemsetAsync(y1, 0, NY1 * sizeof(float), stream);
    hipMemsetAsync(y2, 0, NIMG * sizeof(float), stream);

    float* xb = x0;
    float* xa = x1;
    const int ebBlocks = (int)(NIMG / 256);                // 1728
    const int pxBlocks = (int)((MBATCH * MM * NNW) / 256); // 576

    for (int k = 0; k < KITER; ++k) {
        k_conv_csc<<<dim3(12, 12, MBATCH * CCH), 256, 0, stream>>>(a, Bw, xb, r);
        k_xupd<<<pxBlocks, 256, 0, stream>>>(xb, r, y1, y2, mask, gam1, k, xa, u);
        k_convT_wmma<<<dim3(12, 24, MBATCH * CCH), 256, 0, stream>>>(r, Bw, a, gam2, lam1, k);
        k_y1<<<pxBlocks, 256, 0, stream>>>(y1, u, gam3, lam2, k);
        k_y2acc<<<ebBlocks, 256, 0, stream>>>(y2, mask, xa, xb, z, gam3, k, part);
        k_y2red<<<MBATCH, 256, 0, stream>>>(part, fac, radius);
        k_y2app<<<ebBlocks, 256, 0, stream>>>(y2, z, fac, gam3, k);
        float* t = xb; xb = xa; xa = t;
    }
    // xb now holds the final x_aft
    hipMemcpyAsync(d_out, xb, NIMG * sizeof(float), hipMemcpyDeviceToDevice, stream);
}